// ProteinFeatures_858993459533
// MI455X (gfx1250) — compile-verified
//
#include <hip/hip_runtime.h>
#include <hip/hip_bf16.h>
#include <math.h>

#define Bn 4
#define Nn 2048
#define Kn 30
#define Cn 128
#define KPAD 40
#define NK (Nn*Kn)
#define V_SIZE ((size_t)Bn*Nn*Cn)
#define E_SIZE ((size_t)Bn*Nn*Kn*Cn)

typedef __attribute__((ext_vector_type(2))) float v2f;
typedef __attribute__((ext_vector_type(8))) float v8f;

struct F3 { float x, y, z; };
__device__ __forceinline__ F3 mk3(float a, float b, float c) { F3 r; r.x=a; r.y=b; r.z=c; return r; }
__device__ __forceinline__ F3 sub3(F3 a, F3 b) { return mk3(a.x-b.x, a.y-b.y, a.z-b.z); }
__device__ __forceinline__ float dot3(F3 a, F3 b) { return a.x*b.x + a.y*b.y + a.z*b.z; }
__device__ __forceinline__ F3 cross3(F3 a, F3 b) {
    return mk3(a.y*b.z - a.z*b.y, a.z*b.x - a.x*b.z, a.x*b.y - a.y*b.x);
}
__device__ __forceinline__ F3 nrm3(F3 a) {
    float n = sqrtf(dot3(a, a));
    float inv = 1.0f / fmaxf(n, 1e-12f);
    return mk3(a.x*inv, a.y*inv, a.z*inv);
}
__device__ __forceinline__ float signf_(float v) { return (v > 0.f) ? 1.f : ((v < 0.f) ? -1.f : 0.f); }
__device__ __forceinline__ F3 loadCA(const float* X, int b, int n) {
    const float* p = X + (((size_t)b*Nn + n)*4 + 1)*3;
    return mk3(p[0], p[1], p[2]);
}

// ---------------------------------------------------------------------------
// Kernel 1: per-node orientation frames O (row-major 3x3: o1, n2, o1 x n2)
// ---------------------------------------------------------------------------
__global__ void pf_frames_kernel(const float* __restrict__ X, float* __restrict__ Ofr) {
    int idx = blockIdx.x * blockDim.x + threadIdx.x;
    if (idx >= Bn*Nn) return;
    int b = idx / Nn, n = idx % Nn;
    float O[9] = {0.f,0.f,0.f,0.f,0.f,0.f,0.f,0.f,0.f};
    if (n >= 1 && n <= Nn - 3) {
        int r = n - 1;
        F3 p0 = loadCA(X, b, r), p1 = loadCA(X, b, r+1), p2 = loadCA(X, b, r+2);
        F3 u2 = nrm3(sub3(p1, p0));
        F3 u1 = nrm3(sub3(p2, p1));
        F3 n2 = nrm3(cross3(u2, u1));
        F3 o1 = nrm3(sub3(u2, u1));
        F3 cx = cross3(o1, n2);
        O[0]=o1.x; O[1]=o1.y; O[2]=o1.z;
        O[3]=n2.x; O[4]=n2.y; O[5]=n2.z;
        O[6]=cx.x; O[7]=cx.y; O[8]=cx.z;
    }
    float* dst = Ofr + (size_t)idx * 9;
    #pragma unroll
    for (int q = 0; q < 9; ++q) dst[q] = O[q];
}

// ---------------------------------------------------------------------------
// Kernel 2: distances + top-K (one block per (b,i))
// ---------------------------------------------------------------------------
__global__ void __launch_bounds__(256) pf_topk_kernel(
        const float* __restrict__ X, const float* __restrict__ mask,
        int* __restrict__ eidx, float* __restrict__ dnb, float* __restrict__ eidx_f) {
    __shared__ float dist[Nn];
    __shared__ float redv[256];
    __shared__ int   redi[256];
    int b = blockIdx.x / Nn, i = blockIdx.x % Nn;
    int tid = threadIdx.x;
    F3 xi = loadCA(X, b, i);
    float mi = mask[b*Nn + i];

    float lmax = -1e30f;
    for (int j = tid; j < Nn; j += 256) {
        F3 d = sub3(xi, loadCA(X, b, j));
        float dv = sqrtf(dot3(d, d) + 1e-6f);
        float m2 = mi * mask[b*Nn + j];
        float v = m2 * dv;
        dist[j] = v;
        lmax = fmaxf(lmax, v);
    }
    redv[tid] = lmax;
    __syncthreads();
    for (int s = 128; s > 0; s >>= 1) {
        if (tid < s) redv[tid] = fmaxf(redv[tid], redv[tid + s]);
        __syncthreads();
    }
    float Dmax = redv[0];
    __syncthreads();
    for (int j = tid; j < Nn; j += 256) {
        float m2 = mi * mask[b*Nn + j];
        dist[j] += (1.0f - m2) * Dmax;
    }
    __syncthreads();

    for (int t = 0; t < Kn; ++t) {
        float bv = 1e30f; int bi = 0x7FFFFFFF;
        for (int j = tid; j < Nn; j += 256) {
            float v = dist[j];
            if (v < bv) { bv = v; bi = j; }
        }
        redv[tid] = bv; redi[tid] = bi;
        __syncthreads();
        for (int s = 128; s > 0; s >>= 1) {
            if (tid < s) {
                float v2 = redv[tid + s]; int i2 = redi[tid + s];
                if (v2 < redv[tid] || (v2 == redv[tid] && i2 < redi[tid])) {
                    redv[tid] = v2; redi[tid] = i2;
                }
            }
            __syncthreads();
        }
        if (tid == 0) {
            int sel = redi[0];
            size_t eo = ((size_t)(b*Nn + i))*Kn + t;
            eidx[eo] = sel;
            dnb[eo] = redv[0];
            eidx_f[eo] = (float)sel;
            dist[sel] = 1e30f;
        }
        __syncthreads();
    }
}

// ---------------------------------------------------------------------------
// Kernel 3: node dihedral features -> 6x128 GEMM -> layernorm (block per node)
// ---------------------------------------------------------------------------
__global__ void __launch_bounds__(128) pf_node_kernel(
        const float* __restrict__ X, const float* __restrict__ W,
        const float* __restrict__ bvec, const float* __restrict__ gain,
        const float* __restrict__ bias, float* __restrict__ Vout) {
    __shared__ float feat[6];
    __shared__ float red[128];
    int bn = blockIdx.x;
    int b = bn / Nn, n = bn % Nn;
    int tid = threadIdx.x;

    if (tid < 3) {
        int s = 3*n + tid;
        float Dv = 0.0f;
        if (s >= 1 && s <= 3*Nn - 3) {
            int t = s - 1;
            F3 p[4];
            #pragma unroll
            for (int q = 0; q < 4; ++q) {
                int pp = t + q;
                const float* src = X + (((size_t)b*Nn + pp/3)*4 + (pp%3))*3;
                p[q] = mk3(src[0], src[1], src[2]);
            }
            F3 u2 = nrm3(sub3(p[1], p[0]));
            F3 u1 = nrm3(sub3(p[2], p[1]));
            F3 u0 = nrm3(sub3(p[3], p[2]));
            F3 n2 = nrm3(cross3(u2, u1));
            F3 n1 = nrm3(cross3(u1, u0));
            float cosD = fminf(fmaxf(dot3(n2, n1), -1.0f + 1e-7f), 1.0f - 1e-7f);
            Dv = signf_(dot3(u2, n1)) * acosf(cosD);
        }
        feat[tid]     = cosf(Dv);
        feat[tid + 3] = sinf(Dv);
    }
    __syncthreads();

    float x = bvec[tid];
    #pragma unroll
    for (int k = 0; k < 6; ++k) x += feat[k] * W[k*Cn + tid];

    red[tid] = x;
    __syncthreads();
    for (int s = 64; s > 0; s >>= 1) { if (tid < s) red[tid] += red[tid + s]; __syncthreads(); }
    float mu = red[0] * (1.0f / 128.0f);
    __syncthreads();
    float dd = x - mu;
    red[tid] = dd * dd;
    __syncthreads();
    for (int s = 64; s > 0; s >>= 1) { if (tid < s) red[tid] += red[tid + s]; __syncthreads(); }
    float var = red[0] * (1.0f / 127.0f);
    float sigma = sqrtf(var + 1e-6f);
    Vout[(size_t)bn*Cn + tid] = gain[tid] * dd / (sigma + 1e-6f) + bias[tid];
}

// ---------------------------------------------------------------------------
// Kernel 4: edge features (LDS) -> fp32 WMMA GEMM 39x128 -> layernorm -> E
// One wave per 16-edge M-tile; 8 waves per block; K padded 39->40 (10 steps
// of V_WMMA_F32_16X16X4_F32), N = 128 (8 tiles of 16).
// ---------------------------------------------------------------------------
__global__ void __launch_bounds__(256) pf_edge_kernel(
        const float* __restrict__ X, const float* __restrict__ W,
        const float* __restrict__ bvec, const float* __restrict__ gain,
        const float* __restrict__ bias,
        const int* __restrict__ eidx, const float* __restrict__ dnb,
        const float* __restrict__ Ofr, float* __restrict__ Eout) {
    __shared__ float Wl[KPAD * Cn];     // 40x128 weights (rows >= 39 zero)
    __shared__ float Fl[8 * 16 * KPAD]; // per-wave 16x40 feature tiles

    int tid = threadIdx.x;
    for (int idx = tid; idx < KPAD * Cn; idx += 256) {
        int k = idx >> 7, nc = idx & 127;
        Wl[idx] = (k < 39) ? W[k*Cn + nc] : 0.0f;
    }

    int wave = tid >> 5, lane = tid & 31;
    int half = lane >> 4, nl = lane & 15;
    int tile = blockIdx.x * 8 + wave;
    float* F = Fl + wave * 16 * KPAD;

    if (lane < 16) {
        int e = tile * 16 + lane;
        int b = e / NK; int rem = e % NK; int i = rem / Kn;
        int j = eidx[e];
        float Dn = dnb[e];
        float f[KPAD];

        // positional embeddings (8 cos, 8 sin)
        float d = (float)(j - i);
        #pragma unroll
        for (int m = 0; m < 8; ++m) {
            float freq = expf((float)(2*m) * (-9.210340371976184f / 16.0f));
            float ang = d * freq;
            f[m]     = cosf(ang);
            f[8 + m] = sinf(ang);
        }
        // RBF (16)
        #pragma unroll
        for (int r = 0; r < 16; ++r) {
            float muv = 20.0f * (float)r / 15.0f;
            float t = (Dn - muv) * (1.0f / 1.25f);
            f[16 + r] = expf(-t * t);
        }
        // direction dU (3) and quaternion (4)
        F3 xi = loadCA(X, b, i), xj = loadCA(X, b, j);
        F3 dXe = sub3(xj, xi);
        const float* Oi = Ofr + ((size_t)b*Nn + i)*9;
        const float* Oj = Ofr + ((size_t)b*Nn + j)*9;
        F3 v = mk3(Oi[0]*dXe.x + Oi[1]*dXe.y + Oi[2]*dXe.z,
                   Oi[3]*dXe.x + Oi[4]*dXe.y + Oi[5]*dXe.z,
                   Oi[6]*dXe.x + Oi[7]*dXe.y + Oi[8]*dXe.z);
        v = nrm3(v);
        f[32] = v.x; f[33] = v.y; f[34] = v.z;

        float R[3][3];
        #pragma unroll
        for (int ii = 0; ii < 3; ++ii)
            #pragma unroll
            for (int l = 0; l < 3; ++l)
                R[ii][l] = Oi[0*3+ii]*Oj[0*3+l] + Oi[1*3+ii]*Oj[1*3+l] + Oi[2*3+ii]*Oj[2*3+l];
        float Rxx = R[0][0], Ryy = R[1][1], Rzz = R[2][2];
        float mx = 0.5f * sqrtf(fabsf(1.0f + Rxx - Ryy - Rzz));
        float my = 0.5f * sqrtf(fabsf(1.0f - Rxx + Ryy - Rzz));
        float mz = 0.5f * sqrtf(fabsf(1.0f - Rxx - Ryy + Rzz));
        float qx = signf_(R[2][1] - R[1][2]) * mx;
        float qy = signf_(R[0][2] - R[2][0]) * my;
        float qz = signf_(R[1][0] - R[0][1]) * mz;
        float qw = 0.5f * sqrtf(fmaxf(1.0f + Rxx + Ryy + Rzz, 0.0f));
        float qn = sqrtf(qx*qx + qy*qy + qz*qz + qw*qw);
        float qi = 1.0f / fmaxf(qn, 1e-12f);
        f[35] = qx*qi; f[36] = qy*qi; f[37] = qz*qi; f[38] = qw*qi;
        f[39] = 0.0f;

        #pragma unroll
        for (int k = 0; k < KPAD; ++k) F[lane*KPAD + k] = f[k];
    }
    __syncthreads();

    // WMMA: A(16x4 fp32) lane = 16*half + m, vgpr holds K = 4s + 2*half + {0,1}
    v8f acc[8] = {};
    for (int s = 0; s < 10; ++s) {
        int k0 = 4*s + 2*half;
        v2f a;
        a.x = F[nl*KPAD + k0];
        a.y = F[nl*KPAD + k0 + 1];
        #pragma unroll
        for (int t = 0; t < 8; ++t) {
            v2f bb;
            bb.x = Wl[k0*Cn + 16*t + nl];
            bb.y = Wl[(k0 + 1)*Cn + 16*t + nl];
            acc[t] = __builtin_amdgcn_wmma_f32_16x16x4_f32(
                false, a, false, bb, (short)0, acc[t], false, false);
        }
    }

    // bias add (per output column)
    #pragma unroll
    for (int t = 0; t < 8; ++t) {
        float bv = bvec[16*t + nl];
        #pragma unroll
        for (int r = 0; r < 8; ++r) acc[t][r] += bv;
    }

    // layernorm per row m = r + 8*half; 128 cols = 8 tiles x 16 lanes of half
    #pragma unroll
    for (int r = 0; r < 8; ++r) {
        float s1 = 0.0f, s2 = 0.0f;
        #pragma unroll
        for (int t = 0; t < 8; ++t) { float vv = acc[t][r]; s1 += vv; s2 += vv*vv; }
        #pragma unroll
        for (int m = 1; m < 16; m <<= 1) {
            s1 += __shfl_xor(s1, m, 32);
            s2 += __shfl_xor(s2, m, 32);
        }
        float mu = s1 * (1.0f / 128.0f);
        float var = fmaxf((s2 - 128.0f * mu * mu) * (1.0f / 127.0f), 0.0f);
        float inv = 1.0f / (sqrtf(var + 1e-6f) + 1e-6f);
        size_t e = (size_t)tile * 16 + r + 8*half;
        float* dst = Eout + e * Cn;
        #pragma unroll
        for (int t = 0; t < 8; ++t) {
            int nc = 16*t + nl;
            dst[nc] = gain[nc] * (acc[t][r] - mu) * inv + bias[nc];
        }
    }
}

// ---------------------------------------------------------------------------
extern "C" void kernel_launch(void* const* d_in, const int* in_sizes, int n_in,
                              void* d_out, int out_size, void* d_ws, size_t ws_size,
                              hipStream_t stream) {
    const float* X      = (const float*)d_in[0];
    const float* mask   = (const float*)d_in[1];
    const float* node_W = (const float*)d_in[2];
    const float* node_b = (const float*)d_in[3];
    const float* edge_W = (const float*)d_in[4];
    const float* edge_b = (const float*)d_in[5];
    const float* gain_n = (const float*)d_in[6];
    const float* bias_n = (const float*)d_in[7];
    const float* gain_e = (const float*)d_in[8];
    const float* bias_e = (const float*)d_in[9];
    // d_in[10] (L) unused by the reference computation

    float* out      = (float*)d_out;
    float* V_out    = out;
    float* E_out    = out + V_SIZE;
    float* Eidx_out = out + V_SIZE + E_SIZE;

    const size_t nEdges = (size_t)Bn * Nn * Kn;   // 245760
    char*  ws      = (char*)d_ws;
    int*   ws_eidx = (int*)ws;
    float* ws_dnb  = (float*)(ws + nEdges * sizeof(int));
    float* ws_O    = (float*)(ws + 2 * nEdges * sizeof(int));

    pf_frames_kernel<<<(Bn*Nn + 255)/256, 256, 0, stream>>>(X, ws_O);
    pf_topk_kernel<<<Bn*Nn, 256, 0, stream>>>(X, mask, ws_eidx, ws_dnb, Eidx_out);
    pf_node_kernel<<<Bn*Nn, 128, 0, stream>>>(X, node_W, node_b, gain_n, bias_n, V_out);
    // 245760 edges / 16 rows per tile / 8 waves per block = 1920 blocks
    pf_edge_kernel<<<1920, 256, 0, stream>>>(X, edge_W, edge_b, gain_e, bias_e,
                                             ws_eidx, ws_dnb, ws_O, E_out);
}